// QuantumLayer_65481071399026
// MI455X (gfx1250) — compile-verified
//
#include <hip/hip_runtime.h>
#include <math.h>

typedef __attribute__((ext_vector_type(2))) float v2f;
typedef __attribute__((ext_vector_type(8))) float v8f;

#define LDS_STRIDE 20  // dwords/row: padded so transpose stores+b64 loads are bank-conflict-free

// ---------------------------------------------------------------------------
// Kernel 1: build the fixed 16x16 variational-block matrix U into d_ws.
// Lane n pushes basis vector |n> through RY(params)+CNOT ring; column n = result.
// U[k*16+n] = <k| Block |n>
// ---------------------------------------------------------------------------
__global__ void build_unitary(const float* __restrict__ params, float* __restrict__ U) {
  const int n = threadIdx.x;
  if (n >= 16) return;
  float x[16];
#pragma unroll
  for (int k = 0; k < 16; ++k) x[k] = (k == n) ? 1.0f : 0.0f;
#pragma unroll
  for (int d = 0; d < 2; ++d) {
    // RY on each wire (qubit i <-> bit (3-i) of flat index)
#pragma unroll
    for (int i = 0; i < 4; ++i) {
      const float th = 0.5f * params[d * 4 + i];
      const float c = __cosf(th), s = __sinf(th);
      const int tb = 1 << (3 - i);
#pragma unroll
      for (int k = 0; k < 16; ++k) {
        if (!(k & tb)) {
          const float x0 = x[k], x1 = x[k | tb];
          x[k]      = c * x0 - s * x1;
          x[k | tb] = s * x0 + c * x1;
        }
      }
    }
    // CNOT ring (0,1),(1,2),(2,3),(3,0): for ctrl-bit set, swap target pair
#pragma unroll
    for (int g = 0; g < 4; ++g) {
      const int ctrl = g, tgt = (g + 1) & 3;
      const int cb = 1 << (3 - ctrl), tb = 1 << (3 - tgt);
#pragma unroll
      for (int k = 0; k < 16; ++k) {
        if ((k & cb) && !(k & tb)) {
          const float t = x[k]; x[k] = x[k | tb]; x[k | tb] = t;
        }
      }
    }
  }
#pragma unroll
  for (int k = 0; k < 16; ++k) U[k * 16 + n] = x[k];
}

// ---------------------------------------------------------------------------
// Kernel 2: per wave, 16 samples/iteration.
//   GEMM1: D = S(16x16, rank-1 rows from trig) x U^T      -> 4x wmma_f32_16x16x4_f32
//   square -> LDS transpose (D-layout -> A-layout)
//   GEMM2: out^ = E x Zsigns                              -> 4x wmma_f32_16x16x4_f32
// ---------------------------------------------------------------------------
__global__ __launch_bounds__(256) void qlayer_kernel(const float* __restrict__ patch,
                                                     const float* __restrict__ U,
                                                     float* __restrict__ out,
                                                     int batch) {
  __shared__ float ldsE[8][16 * LDS_STRIDE];  // per-wave transpose buffer
  __shared__ float ldsO[8][64];               // per-wave output staging

  const int lane = threadIdx.x & 31;
  const int w    = threadIdx.x >> 5;
  const int col  = lane & 15;   // N index (GEMM1: amplitude; GEMM2: output qubit)
  const int hi   = lane >> 4;   // lane half <-> K-half inside each K=4 chunk

  // B operand for GEMM1: chunk c holds rows k = 4c + 2*hi + j of U^T, column = col.
  // U^T[k][col] = U[col][k] = U[col*16 + k]; consecutive j -> one b64 load each.
  const v2f* U2 = (const v2f*)U;
  v2f b0 = U2[col * 8 + hi + 0];
  v2f b1 = U2[col * 8 + hi + 2];
  v2f b2 = U2[col * 8 + hi + 4];
  v2f b3 = U2[col * 8 + hi + 6];

  // B operand for GEMM2: Z sign matrix, Zb[k][n] = n<4 ? (bit(3-n) of k ? -1 : +1) : 0
  v2f zb[4];
#pragma unroll
  for (int c = 0; c < 4; ++c) {
#pragma unroll
    for (int j = 0; j < 2; ++j) {
      const int k = 4 * c + 2 * hi + j;
      float v = 0.0f;
      if (col < 4) v = ((k >> (3 - col)) & 1) ? -1.0f : 1.0f;
      zb[c][j] = v;
    }
  }

  const int tiles   = batch >> 4;
  const int wave_id = blockIdx.x * (blockDim.x >> 5) + w;
  const int nwaves  = gridDim.x * (blockDim.x >> 5);

  for (int t = wave_id; t < tiles; t += nwaves) {
    const int base = t << 4;

    // ---- build A (encoded states) in WMMA A-layout ----
    // chunk c, reg j, lane half hi -> k = 4c+2hi+j ; bits: q0=c>>1 q1=c&1 q2=hi q3=j
    const float4 p = ((const float4*)patch)[base + col];
    const float c0 = __cosf(0.5f * p.x), s0 = __sinf(0.5f * p.x);
    const float c1 = __cosf(0.5f * p.y), s1 = __sinf(0.5f * p.y);
    const float c2 = __cosf(0.5f * p.z), s2 = __sinf(0.5f * p.z);
    const float c3 = __cosf(0.5f * p.w), s3 = __sinf(0.5f * p.w);
    const float g2 = hi ? s2 : c2;
    const float q0 = c0 * g2, q1 = s0 * g2;
    const float p0 = q0 * c1, p1 = q0 * s1, p2 = q1 * c1, p3 = q1 * s1;
    v2f a0 = {p0 * c3, p0 * s3};
    v2f a1 = {p1 * c3, p1 * s3};
    v2f a2 = {p2 * c3, p2 * s3};
    v2f a3 = {p3 * c3, p3 * s3};

    // ---- GEMM1: D[b][j] = sum_k S[b][k] * U[j][k] ----
    v8f acc = {};
    acc = __builtin_amdgcn_wmma_f32_16x16x4_f32(false, a0, false, b0, (short)0, acc, false, false);
    acc = __builtin_amdgcn_wmma_f32_16x16x4_f32(false, a1, false, b1, (short)0, acc, false, false);
    acc = __builtin_amdgcn_wmma_f32_16x16x4_f32(false, a2, false, b2, (short)0, acc, false, false);
    acc = __builtin_amdgcn_wmma_f32_16x16x4_f32(false, a3, false, b3, (short)0, acc, false, false);

    // amplitudes -> probabilities
    v8f e;
#pragma unroll
    for (int v = 0; v < 8; ++v) e[v] = acc[v] * acc[v];

    // ---- transpose C/D-layout (lane=amplitude) -> A-layout (lane=sample) via LDS ----
#pragma unroll
    for (int v = 0; v < 8; ++v)
      ldsE[w][(v + 8 * hi) * LDS_STRIDE + col] = e[v];
    asm volatile("s_wait_dscnt 0x0" ::: "memory");  // in-wave DS ordering; no block barrier needed

    v2f e0 = *(const v2f*)&ldsE[w][col * LDS_STRIDE + 0  + 2 * hi];
    v2f e1 = *(const v2f*)&ldsE[w][col * LDS_STRIDE + 4  + 2 * hi];
    v2f e2 = *(const v2f*)&ldsE[w][col * LDS_STRIDE + 8  + 2 * hi];
    v2f e3 = *(const v2f*)&ldsE[w][col * LDS_STRIDE + 12 + 2 * hi];

    // ---- GEMM2: out[b][i] = sum_k E[b][k] * z_i(k) ----
    v8f acc2 = {};
    acc2 = __builtin_amdgcn_wmma_f32_16x16x4_f32(false, e0, false, zb[0], (short)0, acc2, false, false);
    acc2 = __builtin_amdgcn_wmma_f32_16x16x4_f32(false, e1, false, zb[1], (short)0, acc2, false, false);
    acc2 = __builtin_amdgcn_wmma_f32_16x16x4_f32(false, e2, false, zb[2], (short)0, acc2, false, false);
    acc2 = __builtin_amdgcn_wmma_f32_16x16x4_f32(false, e3, false, zb[3], (short)0, acc2, false, false);

    // ---- stage the 4 valid output columns, then fully coalesced b64 store ----
    if (col < 4) {
#pragma unroll
      for (int v = 0; v < 8; ++v)
        ldsO[w][(v + 8 * hi) * 4 + col] = acc2[v];
    }
    asm volatile("s_wait_dscnt 0x0" ::: "memory");
    const v2f o = ((const v2f*)ldsO[w])[lane];
    ((v2f*)(out + (size_t)base * 4))[lane] = o;
  }
}

// ---------------------------------------------------------------------------
extern "C" void kernel_launch(void* const* d_in, const int* in_sizes, int n_in,
                              void* d_out, int out_size, void* d_ws, size_t ws_size,
                              hipStream_t stream) {
  const float* patch  = (const float*)d_in[0];   // (B, 4) float32
  const float* params = (const float*)d_in[1];   // (2, 4) float32
  float* out = (float*)d_out;                    // (B, 4) float32
  float* U   = (float*)d_ws;                     // 256 floats scratch

  const int batch = in_sizes[0] / 4;

  build_unitary<<<1, 32, 0, stream>>>(params, U);

  const int tiles = batch >> 4;                  // 16 samples per wave-iteration
  const int waves_per_block = 256 / 32;
  int blocks = (tiles + waves_per_block - 1) / waves_per_block;
  if (blocks > 1024) blocks = 1024;              // grid-stride; ~8 iters/wave at B=1M
  if (blocks < 1) blocks = 1;
  qlayer_kernel<<<blocks, 256, 0, stream>>>(patch, U, out, batch);
}